// CustomMultiTaskLoss_69655779607183
// MI455X (gfx1250) — compile-verified
//
#include <hip/hip_runtime.h>
#include <hip/hip_bf16.h>

typedef __attribute__((ext_vector_type(2))) float v2f;
typedef __attribute__((ext_vector_type(8))) float v8f;

#define WBL 0.5f

__device__ __forceinline__ void atomAddF(float* p, float v) {
    // Lowers to global_atomic_add_f32 (no CAS loop) on gfx1250.
    unsafeAtomicAdd(p, v);
}

__device__ __forceinline__ v8f wmma_acc(v2f a, v2f b, v8f c) {
    return __builtin_amdgcn_wmma_f32_16x16x4_f32(
        /*neg_a=*/false, a, /*neg_b=*/false, b,
        /*c_mod=*/(short)0, c, /*reuse_a=*/false, /*reuse_b=*/false);
}

// Reduce the wave's WMMA accumulator (8 VGPRs x 32 lanes = 256 f32) to one
// scalar and atomically add sum/16 (ones-matrix A replicates column sums 16x).
__device__ __forceinline__ void wave_commit(v8f c, int lane, float* target) {
    float s = c[0] + c[1] + c[2] + c[3] + c[4] + c[5] + c[6] + c[7];
    for (int off = 16; off > 0; off >>= 1)
        s += __shfl_xor(s, off, 32);
    if (lane == 0)
        atomAddF(target, s * (1.0f / 16.0f));
}

// Fused: d = pred - in, acc = 0, and WMMA-reduce sum(|pred - gt|) -> sums[0].
__global__ void k_prep(const float* __restrict__ pred,
                       const float* __restrict__ gt,
                       const float* __restrict__ inp,
                       float* __restrict__ d,
                       float* __restrict__ acc,
                       float* __restrict__ sumL1,
                       int n3) {
    const int lane = threadIdx.x & 31;
    // Force wave id into an SGPR so the chunk loop is a scalar loop.
    const int wave =
        __builtin_amdgcn_readfirstlane((blockIdx.x * blockDim.x + threadIdx.x) >> 5);
    const int stride = ((gridDim.x * blockDim.x) >> 5) * 64;

    v2f a; a.x = 1.0f; a.y = 1.0f;   // ones matrix: D row = column sums of B
    v8f c = {};

    int base = wave * 64;
    // Main body: full 64-element chunks, vectorized b64 traffic, no guards.
    for (; base + 64 <= n3; base += stride) {
        const int i0 = base + 2 * lane;
        float2 p = *(const float2*)(pred + i0);
        float2 g = *(const float2*)(gt + i0);
        float2 x = *(const float2*)(inp + i0);
        float2 dd; dd.x = p.x - x.x; dd.y = p.y - x.y;
        *(float2*)(d + i0) = dd;
        float2 z; z.x = 0.0f; z.y = 0.0f;
        *(float2*)(acc + i0) = z;
        v2f b; b.x = fabsf(p.x - g.x); b.y = fabsf(p.y - g.y);
        c = wmma_acc(a, b, c);
    }
    // Tail chunk: uniform per-wave branch; guards close before the WMMA.
    if (base < n3) {
        const int i0 = base + 2 * lane;
        const int i1 = i0 + 1;
        float t0 = 0.0f, t1 = 0.0f;
        if (i0 < n3) {
            float p = pred[i0], g = gt[i0], x = inp[i0];
            d[i0] = p - x; acc[i0] = 0.0f; t0 = fabsf(p - g);
        }
        if (i1 < n3) {
            float p = pred[i1], g = gt[i1], x = inp[i1];
            d[i1] = p - x; acc[i1] = 0.0f; t1 = fabsf(p - g);
        }
        v2f b; b.x = t0; b.y = t1;
        c = wmma_acc(a, b, c);
    }
    wave_commit(c, lane, sumL1);
}

// One thread per edge: acc[dst] += d[dst] - d[src], per component.
// Index stream is sequential; d/acc (1.2 MB) are L2-resident.
__global__ void k_edges(const int* __restrict__ src,
                        const int* __restrict__ dst,
                        const float* __restrict__ d,
                        float* __restrict__ acc,
                        int E) {
    const int e = blockIdx.x * blockDim.x + threadIdx.x;
    if (e >= E) return;
    if (e + 16384 < E) {           // pull index stream ahead of the atomics
        __builtin_prefetch(src + e + 16384, 0, 0);  // global_prefetch_b8
        __builtin_prefetch(dst + e + 16384, 0, 0);
    }
    const int s3 = src[e] * 3;
    const int t3 = dst[e] * 3;
    const float dx = d[t3]     - d[s3];
    const float dy = d[t3 + 1] - d[s3 + 1];
    const float dz = d[t3 + 2] - d[s3 + 2];
    atomAddF(acc + t3,     dx);
    atomAddF(acc + t3 + 1, dy);
    atomAddF(acc + t3 + 2, dz);
}

// WMMA-reduce sum(|acc|) -> sums[1].
__global__ void k_lapsum(const float* __restrict__ acc,
                         float* __restrict__ sumLap,
                         int n3) {
    const int lane = threadIdx.x & 31;
    const int wave =
        __builtin_amdgcn_readfirstlane((blockIdx.x * blockDim.x + threadIdx.x) >> 5);
    const int stride = ((gridDim.x * blockDim.x) >> 5) * 64;

    v2f a; a.x = 1.0f; a.y = 1.0f;
    v8f c = {};

    int base = wave * 64;
    for (; base + 64 <= n3; base += stride) {
        const int i0 = base + 2 * lane;
        float2 x = *(const float2*)(acc + i0);
        v2f b; b.x = fabsf(x.x); b.y = fabsf(x.y);
        c = wmma_acc(a, b, c);
    }
    if (base < n3) {
        const int i0 = base + 2 * lane;
        const int i1 = i0 + 1;
        v2f b;
        b.x = (i0 < n3) ? fabsf(acc[i0]) : 0.0f;
        b.y = (i1 < n3) ? fabsf(acc[i1]) : 0.0f;
        c = wmma_acc(a, b, c);
    }
    wave_commit(c, lane, sumLap);
}

__global__ void k_init(float* sums) {
    sums[0] = 0.0f;
    sums[1] = 0.0f;
}

__global__ void k_final(const float* __restrict__ sums,
                        float* __restrict__ out,
                        float inv_count) {
    const float l1  = sums[0] * inv_count;
    const float lap = sums[1] * inv_count;
    out[0] = l1 + WBL * lap;
    out[1] = l1;
    out[2] = lap;
}

extern "C" void kernel_launch(void* const* d_in, const int* in_sizes, int n_in,
                              void* d_out, int out_size, void* d_ws, size_t ws_size,
                              hipStream_t stream) {
    const float* pred = (const float*)d_in[0];
    const float* gt   = (const float*)d_in[1];
    const float* inp  = (const float*)d_in[2];
    const int*   ei   = (const int*)d_in[3];   // edge_index flat [2, E]

    const int n3 = in_sizes[0];        // N * 3
    const int E  = in_sizes[3] / 2;
    const int* src = ei;
    const int* dst = ei + E;

    float* ws   = (float*)d_ws;
    float* sums = ws;                  // [0]=sumL1, [1]=sumLap
    float* d    = ws + 2;              // n3 floats (pred - in)
    float* acc  = ws + 2 + n3;         // n3 floats (Laplacian accumulator)
    float* out  = (float*)d_out;

    k_init<<<1, 1, 0, stream>>>(sums);

    // Fixed 1024-wave grid (128 blocks x 8 waves), grid-stride over 64-elem chunks.
    const int rblocks = 128;
    k_prep<<<rblocks, 256, 0, stream>>>(pred, gt, inp, d, acc, sums + 0, n3);

    k_edges<<<(E + 255) / 256, 256, 0, stream>>>(src, dst, d, acc, E);

    k_lapsum<<<rblocks, 256, 0, stream>>>(acc, sums + 1, n3);

    k_final<<<1, 1, 0, stream>>>(sums, out, 1.0f / (float)n3);
}